// HardTripletloss_73564199846203
// MI455X (gfx1250) — compile-verified
//
#include <hip/hip_runtime.h>
#include <hip/hip_bf16.h>
#include <math.h>

typedef float v2f __attribute__((ext_vector_type(2)));
typedef float v8f __attribute__((ext_vector_type(8)));

#define MARGIN 0.3f
#define EPS    1e-8f
#define NROWS  49

// ---------------------------------------------------------------------------
// Kernel 1: per-(group, column-block) partial Gram via V_WMMA_F32_16X16X4_F32.
// Group g holds 16 rows: slot 0 = anchor (row 0), slot s>=1 = row g*15+s
// (clamped to 48; padded slots are ignored by the finalize kernel).
// A(16x4) and B(4x16) operand layouts for Gram = X * X^T coincide:
//   lane L (L<16):   V0 = x_L[c+0], V1 = x_L[c+1]
//   lane L+16:       V0 = x_L[c+2], V1 = x_L[c+3]
// so a single global_load_b64 per lane feeds both WMMA operands.
// Four independent accumulation chains -> 4 outstanding loads per wave for
// memory-level parallelism; merged with v_pk_add_f32 at the end.
// Accumulator D(16x16) f32 layout: VGPR j, lanes 0-15 -> M=j, N=lane;
// lanes 16-31 -> M=j+8, N=lane-16.
//   dots (M=0 row):       acc[0] on lanes 0..15
//   norms m=0..7:         acc[m]   on lane m        (m == lane)
//   norms m=8..15:        acc[m-8] on lane m+16     (lanes 24..31)
// ---------------------------------------------------------------------------
__global__ __launch_bounds__(32) void gram_partial(const float* __restrict__ img,
                                                   float* __restrict__ ws,
                                                   int D, int ncb, int cpb)
{
    const int lane = threadIdx.x;        // 0..31, wave32
    const int g    = blockIdx.y;         // 0..3
    const int cb   = blockIdx.x;         // 0..ncb-1
    const int slot = lane & 15;
    const int koff = (lane >> 4) << 1;   // 0 for lanes 0-15, 2 for lanes 16-31

    int r = (slot == 0) ? 0 : (g * 15 + slot);
    if (r > NROWS - 1) r = NROWS - 1;    // padding slots read valid memory
    const float* rowp = img + (size_t)r * (size_t)D;

    const int c0 = cb * cpb;
    v8f acc0 = {};
    v8f acc1 = {};
    v8f acc2 = {};
    v8f acc3 = {};

    // Four K=4 chunks per iteration -> four independent accumulation chains.
    for (int c = c0; c < c0 + cpb; c += 16) {
        v2f a0 = *(const v2f*)(rowp + c +  0 + koff);
        v2f a1 = *(const v2f*)(rowp + c +  4 + koff);
        v2f a2 = *(const v2f*)(rowp + c +  8 + koff);
        v2f a3 = *(const v2f*)(rowp + c + 12 + koff);
        acc0 = __builtin_amdgcn_wmma_f32_16x16x4_f32(false, a0, false, a0,
                                                     (short)0, acc0, false, false);
        acc1 = __builtin_amdgcn_wmma_f32_16x16x4_f32(false, a1, false, a1,
                                                     (short)0, acc1, false, false);
        acc2 = __builtin_amdgcn_wmma_f32_16x16x4_f32(false, a2, false, a2,
                                                     (short)0, acc2, false, false);
        acc3 = __builtin_amdgcn_wmma_f32_16x16x4_f32(false, a3, false, a3,
                                                     (short)0, acc3, false, false);
    }
    v8f acc = (acc0 + acc1) + (acc2 + acc3);

    float* out = ws + ((size_t)(g * ncb + cb)) * 32;

    // dots: D[0][n] for n = 0..15
    if (lane < 16) out[lane] = acc[0];

    // norms: pick acc[lane & 7] without dynamic vector indexing
    float nv;
    switch (lane & 7) {
        case 0: nv = acc[0]; break;
        case 1: nv = acc[1]; break;
        case 2: nv = acc[2]; break;
        case 3: nv = acc[3]; break;
        case 4: nv = acc[4]; break;
        case 5: nv = acc[5]; break;
        case 6: nv = acc[6]; break;
        default: nv = acc[7]; break;
    }
    if (lane < 8)        out[16 + lane] = nv;   // m = 0..7
    else if (lane >= 24) out[lane]      = nv;   // m = 8..15 -> slots 24..31
}

// ---------------------------------------------------------------------------
// Kernel 2: reduce partials across column blocks, then scalar top-k epilogue.
// ---------------------------------------------------------------------------
__global__ __launch_bounds__(128) void finalize(const float* __restrict__ ws,
                                                float* __restrict__ out,
                                                int ncb)
{
    __shared__ float tot[128];           // [g*32 + i]: i<16 dots, i>=16 norms
    const int t = threadIdx.x;
    const int g = t >> 5;
    const int i = t & 31;
    float s = 0.0f;
    for (int cb = 0; cb < ncb; ++cb)
        s += ws[((size_t)(g * ncb + cb)) * 32 + i];
    tot[t] = s;
    __syncthreads();

    if (t == 0) {
        const float normA = tot[16];                 // group 0, slot 0 norm
        const float na = fmaxf(sqrtf(normA), EPS);

        float dist[NROWS];
        dist[0] = 0.0f;
        for (int gg = 0; gg < 4; ++gg) {
            for (int ss = 1; ss < 16; ++ss) {
                const int r = gg * 15 + ss;
                if (r <= NROWS - 1) {
                    const float dot = tot[gg * 32 + ss];
                    const float nb  = fmaxf(sqrtf(tot[gg * 32 + 16 + ss]), EPS);
                    dist[r] = 1.0f - dot / (na * nb);
                }
            }
        }

        // mean of 4 largest positive distances (rows 1..16)
        float dp[16];
        for (int k = 0; k < 16; ++k) dp[k] = dist[1 + k];
        float sum_p = 0.0f;
        for (int k = 0; k < 4; ++k) {
            int bi = 0; float bv = -1e30f;
            for (int j = 0; j < 16; ++j) if (dp[j] > bv) { bv = dp[j]; bi = j; }
            sum_p += bv;
            dp[bi] = -1e30f;
        }
        const float mean_p = sum_p * 0.25f;

        // 8 smallest negative distances (rows 17..48)
        float dn[32];
        for (int k = 0; k < 32; ++k) dn[k] = dist[17 + k];
        float loss = 0.0f;
        for (int k = 0; k < 8; ++k) {
            int bi = 0; float bv = 1e30f;
            for (int j = 0; j < 32; ++j) if (dn[j] < bv) { bv = dn[j]; bi = j; }
            loss += fmaxf(mean_p - bv + MARGIN, 0.0f);
            dn[bi] = 1e30f;
        }
        out[0] = loss * 0.125f;
    }
}

extern "C" void kernel_launch(void* const* d_in, const int* in_sizes, int n_in,
                              void* d_out, int out_size, void* d_ws, size_t ws_size,
                              hipStream_t stream)
{
    const float* img = (const float*)d_in[0];
    float* out = (float*)d_out;
    float* ws  = (float*)d_ws;

    const int D = in_sizes[0] / NROWS;   // 1048576

    // Choose column-block count: power of two, fits ws (4 groups * 32 floats each),
    // and cpb = D / ncb must be a multiple of 16 (4-deep unrolled WMMA loop).
    int ncb = 512;
    const size_t bytes_per_cb = 4ull * 32ull * sizeof(float);
    while (ncb > 1 && ((size_t)ncb * bytes_per_cb > ws_size || (D / ncb) % 16 != 0))
        ncb >>= 1;
    const int cpb = D / ncb;

    dim3 grid(ncb, 4);
    gram_partial<<<grid, 32, 0, stream>>>(img, ws, D, ncb, cpb);
    finalize<<<1, 128, 0, stream>>>(ws, out, ncb);
}